// LevelPredictor_26104811225562
// MI455X (gfx1250) — compile-verified
//
#include <hip/hip_runtime.h>
#include <hip/hip_bf16.h>

typedef float v2f __attribute__((ext_vector_type(2)));
typedef float v8f __attribute__((ext_vector_type(8)));

// ---------------------------------------------------------------- zero fill
__global__ void zero_f32_kernel(float* __restrict__ p, long long n) {
  long long i = (long long)blockIdx.x * blockDim.x + threadIdx.x;
  long long stride = (long long)gridDim.x * blockDim.x;
  for (; i < n; i += stride) p[i] = 0.0f;
}

// ---------------------------------------------------------------- degree
__global__ void degree_kernel(const long long* __restrict__ dst,
                              float* __restrict__ deg, long long E) {
  long long i = (long long)blockIdx.x * blockDim.x + threadIdx.x;
  long long stride = (long long)gridDim.x * blockDim.x;
  for (; i < E; i += stride) {
    atomicAdd(&deg[dst[i]], 1.0f);
  }
}

// ---------------------------------------------------------------- scatter-add of messages
// work item = (edge, 4-channel chunk). float4 gather from x[src], 4 atomic adds
// into agg[dst]. agg (<=51MB) is L2 resident on MI455X (192MB L2).
__global__ void scatter_add_kernel(const float* __restrict__ x,
                                   const long long* __restrict__ src,
                                   const long long* __restrict__ dst,
                                   float* __restrict__ agg,
                                   long long nwork, int C, int chshift) {
  long long i = (long long)blockIdx.x * blockDim.x + threadIdx.x;
  long long stride = (long long)gridDim.x * blockDim.x;
  const int chmask = (1 << chshift) - 1;
  for (; i < nwork; i += stride) {
    long long e = i >> chshift;
    int ch = (int)(i & chmask) << 2;
    long long s = src[e];
    long long d = dst[e];
    const float4 v = *(const float4*)(x + s * (long long)C + ch);
    float* o = agg + d * (long long)C + ch;
    atomicAdd(o + 0, v.x);
    atomicAdd(o + 1, v.y);
    atomicAdd(o + 2, v.z);
    atomicAdd(o + 3, v.w);
  }
}

// ---------------------------------------------------------------- fused SAGE layer (WMMA fp32)
// out[m, n] = prelu( (agg[m,:]/max(deg,1)) @ Wl + xin[m,:] @ Wr + b )
// One wave per 16x16 output tile; 4 waves (4 N-tiles) per block.
// A fragment (16x4 f32): lane L<16 -> row m0+L, cols kk..kk+1 in v0,v1;
//                        lane L>=16 -> row m0+L-16, cols kk+2..kk+3.
// B fragment (4x16 f32): v0 = W[kk + 2*(L>=16)][n0+(L&15)], v1 = next K row.
__global__ __launch_bounds__(128)
void sage_gemm_wmma_kernel(const float* __restrict__ xin,
                           const float* __restrict__ agg,
                           const float* __restrict__ deg,
                           const float* __restrict__ Wl,
                           const float* __restrict__ Wr,
                           const float* __restrict__ bias,
                           const float* __restrict__ prelu_a,
                           float* __restrict__ out,
                           int C_in, int C_out, int apply_prelu) {
  const int lane = threadIdx.x & 31;
  const int wave = threadIdx.x >> 5;
  const int m0 = blockIdx.x * 16;
  const int n0 = (blockIdx.y * 4 + wave) * 16;

  const int arow = m0 + (lane & 15);      // A-matrix row this lane feeds
  const int koff = (lane >> 4) << 1;      // 0 for lanes 0-15, 2 for 16-31
  const int ncol = n0 + (lane & 15);      // B/D column this lane feeds

  const float invd = 1.0f / fmaxf(deg[arow], 1.0f);

  v8f c = {0.f, 0.f, 0.f, 0.f, 0.f, 0.f, 0.f, 0.f};

  // ---- (agg/deg) @ Wl
  const float* ap = agg + (size_t)arow * C_in;
  for (int kk = 0; kk < C_in; kk += 4) {
    v2f a;
    a.x = ap[kk + koff + 0] * invd;
    a.y = ap[kk + koff + 1] * invd;
    const float* wp = Wl + (size_t)(kk + koff) * C_out + ncol;
    v2f bf;
    bf.x = wp[0];
    bf.y = wp[C_out];
    c = __builtin_amdgcn_wmma_f32_16x16x4_f32(false, a, false, bf,
                                              (short)0, c, false, false);
  }
  // ---- xin @ Wr
  const float* xp = xin + (size_t)arow * C_in;
  for (int kk = 0; kk < C_in; kk += 4) {
    v2f a;
    a.x = xp[kk + koff + 0];
    a.y = xp[kk + koff + 1];
    const float* wp = Wr + (size_t)(kk + koff) * C_out + ncol;
    v2f bf;
    bf.x = wp[0];
    bf.y = wp[C_out];
    c = __builtin_amdgcn_wmma_f32_16x16x4_f32(false, a, false, bf,
                                              (short)0, c, false, false);
  }

  // ---- epilogue: bias + PReLU + store.
  // C/D layout: VGPR v -> M = v (lanes 0-15) or v+8 (lanes 16-31), N = lane&15.
  const float bv = bias[ncol];
  const float alpha = prelu_a[0];
  const int rbase = m0 + ((lane >> 4) << 3);
  for (int v = 0; v < 8; ++v) {
    float val = c[v] + bv;
    if (apply_prelu) val = (val > 0.0f) ? val : alpha * val;
    out[(size_t)(rbase + v) * C_out + ncol] = val;
  }
}

// ---------------------------------------------------------------- scalar tail for rows N - N%16
__global__ void sage_rows_scalar_kernel(const float* __restrict__ xin,
                                        const float* __restrict__ agg,
                                        const float* __restrict__ deg,
                                        const float* __restrict__ Wl,
                                        const float* __restrict__ Wr,
                                        const float* __restrict__ bias,
                                        const float* __restrict__ prelu_a,
                                        float* __restrict__ out,
                                        int C_in, int C_out, int apply_prelu,
                                        int row_start, int nrows) {
  int idx = blockIdx.x * blockDim.x + threadIdx.x;
  if (idx >= nrows * C_out) return;
  int row = row_start + idx / C_out;
  int col = idx % C_out;
  float invd = 1.0f / fmaxf(deg[row], 1.0f);
  float acc = bias[col];
  const float* ap = agg + (size_t)row * C_in;
  const float* xp = xin + (size_t)row * C_in;
  for (int k = 0; k < C_in; ++k) {
    acc += ap[k] * invd * Wl[(size_t)k * C_out + col];
    acc += xp[k] * Wr[(size_t)k * C_out + col];
  }
  if (apply_prelu) acc = (acc > 0.0f) ? acc : prelu_a[0] * acc;
  out[(size_t)row * C_out + col] = acc;
}

// ---------------------------------------------------------------- head: level = h @ W_head + b
__global__ void head_kernel(const float* __restrict__ h,
                            const float* __restrict__ Wh,
                            const float* __restrict__ bh,
                            float* __restrict__ out, int N) {
  const int lane = threadIdx.x & 31;
  const int node = blockIdx.x * (blockDim.x >> 5) + (threadIdx.x >> 5);
  if (node >= N) return;
  const float2 hv = *(const float2*)(h + (size_t)node * 64 + lane * 2);
  const float2 wv = *(const float2*)(Wh + lane * 2);
  float p = hv.x * wv.x + hv.y * wv.y;
  for (int off = 16; off > 0; off >>= 1) p += __shfl_xor(p, off, 32);
  if (lane == 0) out[node] = p + bh[0];
}

// ---------------------------------------------------------------- launcher
extern "C" void kernel_launch(void* const* d_in, const int* in_sizes, int n_in,
                              void* d_out, int out_size, void* d_ws, size_t ws_size,
                              hipStream_t stream) {
  const float* x = (const float*)d_in[0];
  const long long* edges = (const long long*)d_in[1];  // int64 [2, E]
  const float* W1l = (const float*)d_in[2];
  const float* b1  = (const float*)d_in[3];
  const float* W1r = (const float*)d_in[4];
  const float* W2l = (const float*)d_in[5];
  const float* b2  = (const float*)d_in[6];
  const float* W2r = (const float*)d_in[7];
  const float* W3l = (const float*)d_in[8];
  const float* b3  = (const float*)d_in[9];
  const float* W3r = (const float*)d_in[10];
  const float* pa  = (const float*)d_in[11];
  const float* Wh  = (const float*)d_in[12];
  const float* bh  = (const float*)d_in[13];

  const long long N = in_sizes[0] / 64;
  const long long E = in_sizes[1] / 2;
  const long long* srcp = edges;
  const long long* dstp = edges + E;

  float* deg = (float*)d_ws;
  float* agg = deg + N;
  float* hA  = agg + N * 128;
  float* hB  = hA + N * 128;

  const int ZB = 256;
  const int zgrid = 4096;

  // degree (once; graph is shared by all layers)
  zero_f32_kernel<<<zgrid, ZB, 0, stream>>>(deg, N);
  {
    long long blocks = (E + ZB - 1) / ZB;
    if (blocks > 200000) blocks = 200000;
    degree_kernel<<<(unsigned)blocks, ZB, 0, stream>>>(dstp, deg, E);
  }

  auto run_layer = [&](const float* xin, int C_in, int C_out,
                       const float* Wl, const float* Wr, const float* b,
                       float* outp, int prelu) {
    // aggregate
    zero_f32_kernel<<<zgrid, ZB, 0, stream>>>(agg, N * (long long)C_in);
    const int chshift = (C_in == 64) ? 4 : 5;  // C_in/4 chunks per edge
    long long nwork = E << chshift;
    long long blocks = (nwork + ZB - 1) / ZB;
    if (blocks > 200000) blocks = 200000;
    scatter_add_kernel<<<(unsigned)blocks, ZB, 0, stream>>>(
        xin, srcp, dstp, agg, nwork, C_in, chshift);

    // fused mean + dual GEMM + bias + prelu (WMMA fp32)
    long long mt = N / 16;
    if (mt > 0) {
      dim3 grid((unsigned)mt, (unsigned)(C_out / 64), 1);
      sage_gemm_wmma_kernel<<<grid, 128, 0, stream>>>(
          xin, agg, deg, Wl, Wr, b, pa, outp, C_in, C_out, prelu);
    }
    int nrem = (int)(N - mt * 16);
    if (nrem > 0) {
      int work = nrem * C_out;
      sage_rows_scalar_kernel<<<(work + ZB - 1) / ZB, ZB, 0, stream>>>(
          xin, agg, deg, Wl, Wr, b, pa, outp, C_in, C_out, prelu,
          (int)(mt * 16), nrem);
    }
  };

  run_layer(x,  64,  128, W1l, W1r, b1, hA, 1);
  run_layer(hA, 128, 128, W2l, W2r, b2, hB, 1);
  run_layer(hB, 128, 64,  W3l, W3r, b3, hA, 0);

  // head: one wave per node
  {
    long long blocks = (N + 7) / 8;
    head_kernel<<<(unsigned)blocks, 256, 0, stream>>>(hA, Wh, bh, (float*)d_out, (int)N);
  }
}